// ROLAND_3994319585327
// MI455X (gfx1250) — compile-verified
//
#include <hip/hip_runtime.h>

// ROLAND layer: 2 x (GCNConv + GRUCell), fp32 throughout.
// Dense matmuls use V_WMMA_F32_16X16X4_F32 (exact fp32 matrix pipe on CDNA5).
//
// VGPR layouts per CDNA5 ISA 7.12.2 (wave32):
//   A 16x4 : lanes 0-15 -> M=lane, VGPR0=K0,VGPR1=K1 ; lanes 16-31 -> K2,K3
//   B 4x16 : lanes 0-15 -> N=lane, VGPR0=K0,VGPR1=K1 ; lanes 16-31 -> K2,K3
//   C 16x16: VGPR v, lane l -> (M = v + 8*(l>>4), N = l&15)

typedef __attribute__((ext_vector_type(2))) float v2f;
typedef __attribute__((ext_vector_type(8))) float v8f;

#define DIN 128
#define HDIM 64

__device__ __forceinline__ v8f wmma4(v2f a, v2f b, v8f c)
{
    // (neg_a, A, neg_b, B, c_mod, C, reuse_a, reuse_b)
    return __builtin_amdgcn_wmma_f32_16x16x4_f32(
        false, a, false, b, (short)0, c, false, false);
}

// ---------------------------------------------------------------------------
// Conv GEMM: C[N x 64] = A[N x K] * B[K x 64].
// One wave per 16-ROW strip: computes all 4 column tiles so the A tile
// (the large streaming operand) is loaded once per K-step and feeds 4 WMMAs.
// B element (k, col) lives at B[k*bks + col*bns].
// ---------------------------------------------------------------------------
__global__ void gemm_wmma(const float* __restrict__ A, int lda,
                          const float* __restrict__ B, int bks, int bns,
                          float* __restrict__ C, int ldc,
                          int K, int rowTiles)
{
    const int wave = (int)((blockIdx.x * blockDim.x + threadIdx.x) >> 5);
    const int lane = threadIdx.x & 31;
    if (wave >= rowTiles) return;                  // wave-uniform guard (EXEC all-1s)
    const int row0 = wave * 16;
    const int m  = lane & 15;
    const int kh = (lane >> 4) << 1;               // 0 or 2: K sub-offset per half-wave

    v8f acc[4];
#pragma unroll
    for (int t = 0; t < 4; ++t)
        acc[t] = (v8f){0.f, 0.f, 0.f, 0.f, 0.f, 0.f, 0.f, 0.f};

    for (int k0 = 0; k0 < K; k0 += 4) {
        v2f a;
        const float* ap = A + (size_t)(row0 + m) * lda + (k0 + kh);
        a.x = ap[0];
        a.y = ap[1];
#pragma unroll
        for (int t = 0; t < 4; ++t) {
            v2f b;
            const float* bp = B + (size_t)(k0 + kh) * bks + (size_t)(t * 16 + m) * bns;
            b.x = bp[0];
            b.y = bp[bks];
            acc[t] = wmma4(a, b, acc[t]);
        }
    }

    const int n     = lane & 15;
    const int mBase = row0 + ((lane >> 4) << 3);
#pragma unroll
    for (int t = 0; t < 4; ++t)
#pragma unroll
        for (int v = 0; v < 8; ++v)
            C[(size_t)(mBase + v) * ldc + (t * 16 + n)] = acc[t][v];
}

// ---------------------------------------------------------------------------
// Fused 3-gate tile GEMM for the GRU: one pass over K computes the r/z/n gate
// tiles that share the same A tile (X or Hp). W stored [3H, H] row-major and
// used transposed: B(k, col) = W[col*H + k]  ->  contiguous b64 loads.
// ---------------------------------------------------------------------------
__device__ __forceinline__ void wmma_tile3(
    const float* __restrict__ A, const float* __restrict__ W,
    int row0, int col0, int lane, v8f acc[3])
{
    const int m  = lane & 15;
    const int kh = (lane >> 4) << 1;
#pragma unroll
    for (int g = 0; g < 3; ++g)
        acc[g] = (v8f){0.f, 0.f, 0.f, 0.f, 0.f, 0.f, 0.f, 0.f};

    for (int k0 = 0; k0 < HDIM; k0 += 4) {
        v2f a;
        const float* ap = A + (size_t)(row0 + m) * HDIM + (k0 + kh);
        a.x = ap[0];
        a.y = ap[1];
#pragma unroll
        for (int g = 0; g < 3; ++g) {
            v2f b;
            const float* bp = W + (size_t)(g * HDIM + col0 + m) * HDIM + (k0 + kh);
            b.x = bp[0];
            b.y = bp[1];
            acc[g] = wmma4(a, b, acc[g]);
        }
    }
}

// ---------------------------------------------------------------------------
// Degree / normalization kernels (self-loop => deg starts at 1)
// ---------------------------------------------------------------------------
__global__ void init_deg(float* __restrict__ deg, int N)
{
    int i = blockIdx.x * blockDim.x + threadIdx.x;
    if (i < N) deg[i] = 1.0f;
}

__global__ void count_deg(const int* __restrict__ dst, float* __restrict__ deg, int E)
{
    int e = blockIdx.x * blockDim.x + threadIdx.x;
    if (e < E) atomicAdd(&deg[dst[e]], 1.0f);
}

__global__ void inv_sqrt_deg(float* __restrict__ deg, int N)
{
    int i = blockIdx.x * blockDim.x + threadIdx.x;
    if (i < N) deg[i] = rsqrtf(deg[i]);            // deg >= 1 always
}

__global__ void zero_f32(float* __restrict__ p, size_t n)
{
    size_t i = (size_t)blockIdx.x * blockDim.x + threadIdx.x;
    if (i < n) p[i] = 0.0f;
}

// ---------------------------------------------------------------------------
// Edge scatter: one wave per edge, 2 channels per lane (float2 gather, 2 atomics).
// agg[dst] += Xh[src] * dinv[src]*dinv[dst]   (L2-resident gather + atomic RMW)
// ---------------------------------------------------------------------------
__global__ void scatter_edges(const float* __restrict__ Xh,
                              const int* __restrict__ src,
                              const int* __restrict__ dst,
                              const float* __restrict__ dinv,
                              float* __restrict__ agg, int E)
{
    long long gid = (long long)blockIdx.x * blockDim.x + threadIdx.x;
    int e = (int)(gid >> 5);
    if (e >= E) return;
    int c = (int)(gid & 31) * 2;
    int s = src[e];
    int d = dst[e];
    float nrm = dinv[s] * dinv[d];
    const float2 v = *(const float2*)(Xh + (size_t)s * HDIM + c);
    float* ap = agg + (size_t)d * HDIM + c;
    atomicAdd(ap + 0, v.x * nrm);
    atomicAdd(ap + 1, v.y * nrm);
}

// ---------------------------------------------------------------------------
// Conv finalize: h = relu(agg + selfloop(Xh*dinv^2) + bias)
// ---------------------------------------------------------------------------
__global__ void conv_finalize(const float* __restrict__ agg,
                              const float* __restrict__ Xh,
                              const float* __restrict__ dinv,
                              const float* __restrict__ bias,
                              float* __restrict__ h, int N)
{
    size_t gid = (size_t)blockIdx.x * blockDim.x + threadIdx.x;
    if (gid >= (size_t)N * HDIM) return;
    int i = (int)(gid >> 6);
    int c = (int)(gid & 63);
    float di = dinv[i];
    float v = agg[gid] + Xh[gid] * di * di + bias[c];
    h[gid] = fmaxf(v, 0.0f);
}

// ---------------------------------------------------------------------------
// Fused GRU cell: one wave per 16x16 output tile.
//   gi = X  @ w_ih^T + b_ih   (r/z/n gates share the X  A-tile -> wmma_tile3)
//   gh = Hp @ w_hh^T + b_hh   (r/z/n gates share the Hp A-tile -> wmma_tile3)
//   r = sigmoid(gi_r+gh_r); z = sigmoid(gi_z+gh_z)
//   n = tanh(gi_n + r*gh_n);  out = (1-z)*n + z*Hp
// ---------------------------------------------------------------------------
__device__ __forceinline__ float sigm(float x) { return 1.0f / (1.0f + __expf(-x)); }
__device__ __forceinline__ float tanh_fast(float x)
{
    float e = __expf(2.0f * x);
    return (e - 1.0f) / (e + 1.0f);
}

__global__ void gru_wmma(const float* __restrict__ X,
                         const float* __restrict__ Hp,
                         const float* __restrict__ w_ih,
                         const float* __restrict__ w_hh,
                         const float* __restrict__ b_ih,
                         const float* __restrict__ b_hh,
                         float* __restrict__ Out, int rowTiles)
{
    const int wave = (int)((blockIdx.x * blockDim.x + threadIdx.x) >> 5);
    const int lane = threadIdx.x & 31;
    const int rowTile = wave >> 2;                 // HDIM/16 == 4 column tiles
    const int colTile = wave & 3;
    if (rowTile >= rowTiles) return;
    const int row0 = rowTile * 16;
    const int col0 = colTile * 16;
    const int n = lane & 15;

    v8f gi[3], gh[3];
    wmma_tile3(X,  w_ih, row0, col0, lane, gi);
    wmma_tile3(Hp, w_hh, row0, col0, lane, gh);

    const float bir = b_ih[0 * HDIM + col0 + n], bhr = b_hh[0 * HDIM + col0 + n];
    const float biz = b_ih[1 * HDIM + col0 + n], bhz = b_hh[1 * HDIM + col0 + n];
    const float bin = b_ih[2 * HDIM + col0 + n], bhn = b_hh[2 * HDIM + col0 + n];

    const int mBase = row0 + ((lane >> 4) << 3);
#pragma unroll
    for (int v = 0; v < 8; ++v) {
        float r  = sigm(gi[0][v] + bir + gh[0][v] + bhr);
        float z  = sigm(gi[1][v] + biz + gh[1][v] + bhz);
        float nn = tanh_fast(gi[2][v] + bin + r * (gh[2][v] + bhn));
        size_t idx = (size_t)(mBase + v) * HDIM + (col0 + n);
        float hp = Hp[idx];
        Out[idx] = (1.0f - z) * nn + z * hp;
    }
}

// ---------------------------------------------------------------------------
extern "C" void kernel_launch(void* const* d_in, const int* in_sizes, int n_in,
                              void* d_out, int out_size, void* d_ws, size_t ws_size,
                              hipStream_t stream)
{
    const float* x     = (const float*)d_in[0];
    const int*   ei    = (const int*)  d_in[1];
    const float* prev0 = (const float*)d_in[2];
    const float* prev1 = (const float*)d_in[3];
    const float* W1    = (const float*)d_in[4];
    const float* b1    = (const float*)d_in[5];
    const float* W2    = (const float*)d_in[6];
    const float* b2    = (const float*)d_in[7];
    const float* w_ih1 = (const float*)d_in[8];
    const float* w_hh1 = (const float*)d_in[9];
    const float* b_ih1 = (const float*)d_in[10];
    const float* b_hh1 = (const float*)d_in[11];
    const float* w_ih2 = (const float*)d_in[12];
    const float* w_hh2 = (const float*)d_in[13];
    const float* b_ih2 = (const float*)d_in[14];
    const float* b_hh2 = (const float*)d_in[15];

    const int N = in_sizes[0] / DIN;
    const int E = in_sizes[1] / 2;
    const int* src = ei;          // edge_index[0]
    const int* dst = ei + E;      // edge_index[1]

    float* out = (float*)d_out;
    float* h1  = out;                         // [N, H]
    float* h2  = out + (size_t)N * HDIM;      // [N, H]

    float* ws   = (float*)d_ws;
    float* dinv = ws;                               // N      (deg -> d^-1/2)
    float* Xh   = dinv + N;                         // N*H    (gemm out / messages)
    float* agg  = Xh + (size_t)N * HDIM;            // N*H    (scatter target)
    float* hbuf = agg + (size_t)N * HDIM;           // N*H    (conv output)

    const int    rowTiles   = (N + 15) / 16;
    const int    gemmBlocks = (rowTiles * 32 + 255) / 256;            // 1 wave/row strip
    const int    gruBlocks  = (rowTiles * 4 * 32 + 255) / 256;        // 1 wave/16x16 tile
    const size_t NH         = (size_t)N * HDIM;
    const int    nhBlocks   = (int)((NH + 255) / 256);
    const int    edgeBlocks = (int)(((long long)E * 32 + 255) / 256);

    // --- normalization (shared by both convs) ---
    init_deg<<<(N + 255) / 256, 256, 0, stream>>>(dinv, N);
    count_deg<<<(E + 255) / 256, 256, 0, stream>>>(dst, dinv, E);
    inv_sqrt_deg<<<(N + 255) / 256, 256, 0, stream>>>(dinv, N);

    // --- layer 1: GCNConv(x, W1) -> relu -> GRU(prev0) -> h1 ---
    gemm_wmma<<<gemmBlocks, 256, 0, stream>>>(x, DIN, W1, HDIM, 1, Xh, HDIM,
                                              DIN, rowTiles);
    zero_f32<<<nhBlocks, 256, 0, stream>>>(agg, NH);
    scatter_edges<<<edgeBlocks, 256, 0, stream>>>(Xh, src, dst, dinv, agg, E);
    conv_finalize<<<nhBlocks, 256, 0, stream>>>(agg, Xh, dinv, b1, hbuf, N);
    gru_wmma<<<gruBlocks, 256, 0, stream>>>(hbuf, prev0, w_ih1, w_hh1,
                                            b_ih1, b_hh1, h1, rowTiles);

    // --- layer 2: GCNConv(h1, W2) -> relu -> GRU(prev1) -> h2 ---
    gemm_wmma<<<gemmBlocks, 256, 0, stream>>>(h1, HDIM, W2, HDIM, 1, Xh, HDIM,
                                              HDIM, rowTiles);
    zero_f32<<<nhBlocks, 256, 0, stream>>>(agg, NH);
    scatter_edges<<<edgeBlocks, 256, 0, stream>>>(Xh, src, dst, dinv, agg, E);
    conv_finalize<<<nhBlocks, 256, 0, stream>>>(agg, Xh, dinv, b2, hbuf, N);
    gru_wmma<<<gruBlocks, 256, 0, stream>>>(hbuf, prev1, w_ih2, w_hh2,
                                            b_ih2, b_hh2, h2, rowTiles);
}